// ProposalLayer_45930380263991
// MI455X (gfx1250) — compile-verified
//
#include <hip/hip_runtime.h>
#include <cstdint>
#include <cstddef>

// ---------------- problem constants (match reference) ----------------
#define BATCH      16
#define NPTS       16384
#define REG_CH     76          // 12*4 + 1 + 12*2 + 3
#define PRE_NMS    9000
#define POST_NMS   512
#define NMS_THR    0.85f
#define NWORDS     282         // ceil(9000/32); last word has 8 valid bits
#define TWO_PI_F   6.28318530717958647692f
#define PI_F       3.14159265358979323846f

// ---------------- CDNA5 async LDS-DMA helpers (gfx1250) ----------------
// Per-lane gather: each lane copies a dword from its global vaddr to its LDS address in vdst (tracked by ASYNCcnt).
__device__ __forceinline__ void async_gather_f32(unsigned lds_byte, const float* gptr) {
  asm volatile("global_load_async_to_lds_b32 %0, %1, off"
               :: "v"(lds_byte), "v"(gptr)
               : "memory");
}
__device__ __forceinline__ void wait_async0() {
  asm volatile("s_wait_asynccnt 0" ::: "memory");
}
// Generic->LDS byte offset: low 32 bits of a flat LDS pointer are the
// wave-relative LDS address (aperture lives in the high bits).
__device__ __forceinline__ unsigned lds_off(const void* p) {
  return (unsigned)(uintptr_t)p;
}

// ---------------- 1) decode_bbox_target ----------------
__global__ __launch_bounds__(256)
void decode_kernel(const float* __restrict__ reg,
                   const float* __restrict__ xyz,
                   const float* __restrict__ anchor,
                   float* __restrict__ props) {
  int pid = blockIdx.x * blockDim.x + threadIdx.x;   // exact: 1024*256 == B*N
  const float* r = reg + (size_t)pid * REG_CH;
  const float* p = xyz + (size_t)pid * 3;

  // argmax over 12 x-bins / z-bins (first max wins, like jnp.argmax)
  int xb = 0; float xm = r[0];
  #pragma unroll
  for (int i = 1; i < 12; ++i) { float v = r[i];      if (v > xm) { xm = v; xb = i; } }
  int zb = 0; float zm = r[12];
  #pragma unroll
  for (int i = 1; i < 12; ++i) { float v = r[12 + i]; if (v > zm) { zm = v; zb = i; } }

  float pos_x = (float)xb * 0.5f + 0.25f - 3.0f + r[24 + xb] * 0.5f + p[0];
  float pos_z = (float)zb * 0.5f + 0.25f - 3.0f + r[36 + zb] * 0.5f + p[2];
  float pos_y = p[1] + r[48];

  int rb = 0; float rm = r[49];
  #pragma unroll
  for (int i = 1; i < 12; ++i) { float v = r[49 + i]; if (v > rm) { rm = v; rb = i; } }
  const float apc = TWO_PI_F / 12.0f;
  float ry = fmodf((float)rb * apc + r[61 + rb] * (apc * 0.5f), TWO_PI_F);
  if (ry < 0.0f) ry += TWO_PI_F;          // jnp mod: result sign follows divisor
  if (ry > PI_F) ry -= TWO_PI_F;

  float h = r[73] * anchor[0] + anchor[0];
  float w = r[74] * anchor[1] + anchor[1];
  float l = r[75] * anchor[2] + anchor[2];

  float* o = props + (size_t)pid * 7;
  o[0] = pos_x;
  o[1] = pos_y + 0.5f * h;                // props[:,1] += props[:,3]/2
  o[2] = pos_z;
  o[3] = h; o[4] = w; o[5] = l;
  o[6] = ry;
}

// ---------------- 2) per-batch descending bitonic sort in LDS ----------------
__global__ __launch_bounds__(1024)
void sort_kernel(const float* __restrict__ cls,
                 float* __restrict__ s_scores,
                 int*   __restrict__ s_idx) {
  __shared__ float key[NPTS];   // 64 KB
  __shared__ int   val[NPTS];   // 64 KB  (128 KB total, fits 320 KB WGP LDS)
  const int b = blockIdx.x;
  const float* sc = cls + (size_t)b * NPTS;

  for (int i = threadIdx.x; i < NPTS; i += blockDim.x) { key[i] = sc[i]; val[i] = i; }
  __syncthreads();

  for (int k = 2; k <= NPTS; k <<= 1) {
    for (int j = k >> 1; j > 0; j >>= 1) {
      for (int i = threadIdx.x; i < NPTS; i += blockDim.x) {
        int ixj = i ^ j;
        if (ixj > i) {
          bool up = ((i & k) == 0);
          float a = key[i], c = key[ixj];
          bool sw = up ? (a < c) : (a > c);    // descending
          if (sw) {
            key[i] = c; key[ixj] = a;
            int t = val[i]; val[i] = val[ixj]; val[ixj] = t;
          }
        }
      }
      __syncthreads();
    }
  }
  for (int i = threadIdx.x; i < NPTS; i += blockDim.x) {
    s_scores[(size_t)b * NPTS + i] = key[i];
    s_idx  [(size_t)b * NPTS + i] = val[i];
  }
}

// ---------------- 3) chunked greedy NMS + rank compaction ----------------
// Equivalent to the reference fori_loop greedy NMS, restructured as
// wave32-chunk-serial (phase A: in-wave serial greedy over 32 boxes via
// __shfl/__ballot; phase B: block-parallel suppression of all later boxes
// against the <=32 survivors). 564 barriers instead of 9000.
__global__ __launch_bounds__(1024)
void nms_kernel(const float* __restrict__ props,
                const int*   __restrict__ s_idx,
                const float* __restrict__ s_scores,
                float* __restrict__ out_boxes,    // (B,512,7)
                float* __restrict__ out_scores) { // (B,512)
  __shared__ float A0[PRE_NMS];       // cu -> x1
  __shared__ float A1[PRE_NMS];       // cv -> y1
  __shared__ float A2[PRE_NMS];       // w  -> x2
  __shared__ float A3[PRE_NMS];       // l  -> y2
  __shared__ unsigned keepw[NWORDS];
  __shared__ unsigned pref[NWORDS];
  __shared__ float ck0[32], ck1[32], ck2[32], ck3[32];  // chunk box staging

  const int b = blockIdx.x;
  const int* sidx  = s_idx + (size_t)b * NPTS;
  const float* pb  = props + (size_t)b * NPTS * 7;
  const float* ssc = s_scores + (size_t)b * NPTS;

  // Async-gather the 4 BEV source columns of the score-sorted boxes into LDS.
  for (int j = threadIdx.x; j < PRE_NMS; j += blockDim.x) {
    const float* row = pb + (size_t)sidx[j] * 7;
    async_gather_f32(lds_off(&A0[j]), row + 0);   // cu = p[:,0]
    async_gather_f32(lds_off(&A1[j]), row + 2);   // cv = p[:,2]
    async_gather_f32(lds_off(&A2[j]), row + 4);   // w  = p[:,4]
    async_gather_f32(lds_off(&A3[j]), row + 5);   // l  = p[:,5]
  }
  wait_async0();
  __syncthreads();

  // In-place convert to BEV corners: [cu-l/2, cv-w/2, cu+l/2, cv+w/2]
  for (int j = threadIdx.x; j < PRE_NMS; j += blockDim.x) {
    float cu = A0[j], cv = A1[j], w = A2[j], l = A3[j];
    A0[j] = cu - 0.5f * l;
    A1[j] = cv - 0.5f * w;
    A2[j] = cu + 0.5f * l;
    A3[j] = cv + 0.5f * w;
  }
  for (int t = threadIdx.x; t < NWORDS; t += blockDim.x) {
    keepw[t] = (t == NWORDS - 1) ? ((1u << (PRE_NMS - 32 * (NWORDS - 1))) - 1u)
                                 : 0xFFFFFFFFu;
  }
  __syncthreads();

  for (int c = 0; c < NWORDS; ++c) {
    const int base = c * 32;
    const int cnt  = (PRE_NMS - base < 32) ? (PRE_NMS - base) : 32;

    // ---- phase A: wave 0 resolves the serial greedy within this chunk ----
    if (threadIdx.x < 32) {
      const int lane = (int)threadIdx.x;
      const int i = base + lane;
      const bool inr = lane < cnt;
      bool kept = inr && ((keepw[c] >> lane) & 1u);
      float x1 = 0.f, y1 = 0.f, x2 = 0.f, y2 = 0.f;
      if (inr) { x1 = A0[i]; y1 = A1[i]; x2 = A2[i]; y2 = A3[i]; }
      ck0[lane] = x1; ck1[lane] = y1; ck2[lane] = x2; ck3[lane] = y2;
      const float ar = (x2 - x1) * (y2 - y1);
      for (int k = 0; k < cnt; ++k) {
        int kk = __shfl((int)kept, k, 32);       // lane-k keep flag broadcast
        if (kk && lane > k) {
          float bx1 = ck0[k], by1 = ck1[k], bx2 = ck2[k], by2 = ck3[k];
          float xx1 = fmaxf(x1, bx1), yy1 = fmaxf(y1, by1);
          float xx2 = fminf(x2, bx2), yy2 = fminf(y2, by2);
          float inter = fmaxf(xx2 - xx1, 0.0f) * fmaxf(yy2 - yy1, 0.0f);
          float bar   = (bx2 - bx1) * (by2 - by1);
          if (inter / (ar + bar - inter) > NMS_THR) kept = false;
        }
      }
      unsigned word = (unsigned)__ballot(kept);  // wave32: low 32 bits
      if (lane == 0) keepw[c] = word;
    }
    __syncthreads();

    // ---- phase B: all waves suppress later boxes vs the chunk survivors ----
    const unsigned word = keepw[c];
    if (word) {
      for (int j = base + cnt + (int)threadIdx.x; j < PRE_NMS; j += blockDim.x) {
        if (!((keepw[j >> 5] >> (j & 31)) & 1u)) continue;  // already gone
        float jx1 = A0[j], jy1 = A1[j], jx2 = A2[j], jy2 = A3[j];
        float jar = (jx2 - jx1) * (jy2 - jy1);
        bool supp = false;
        for (int k = 0; k < cnt; ++k) {
          if (!((word >> k) & 1u)) continue;
          float bx1 = ck0[k], by1 = ck1[k], bx2 = ck2[k], by2 = ck3[k];
          float xx1 = fmaxf(jx1, bx1), yy1 = fmaxf(jy1, by1);
          float xx2 = fminf(jx2, bx2), yy2 = fminf(jy2, by2);
          float inter = fmaxf(xx2 - xx1, 0.0f) * fmaxf(yy2 - yy1, 0.0f);
          float bar   = (bx2 - bx1) * (by2 - by1);
          if (inter / (jar + bar - inter) > NMS_THR) { supp = true; break; }
        }
        if (supp) atomicAnd(&keepw[j >> 5], ~(1u << (j & 31)));  // ds_and_b32
      }
    }
    __syncthreads();
  }

  // Exclusive popcount prefix over keep words (282 words: trivially serial).
  if (threadIdx.x == 0) {
    unsigned run = 0;
    for (int t = 0; t < NWORDS; ++t) { pref[t] = run; run += __popc(keepw[t]); }
  }
  // Zero-fill this batch's output slice (valid-mask semantics).
  for (int t = threadIdx.x; t < POST_NMS; t += blockDim.x) {
    out_scores[(size_t)b * POST_NMS + t] = 0.0f;
    float* o = out_boxes + ((size_t)b * POST_NMS + t) * 7;
    #pragma unroll
    for (int c2 = 0; c2 < 7; ++c2) o[c2] = 0.0f;
  }
  __syncthreads();

  // Compaction: kept boxes keep their score order -> rank = #kept before j.
  for (int j = threadIdx.x; j < PRE_NMS; j += blockDim.x) {
    unsigned w = keepw[j >> 5];
    if (w & (1u << (j & 31))) {
      int rank = (int)pref[j >> 5] + __popc(w & ((1u << (j & 31)) - 1u));
      if (rank < POST_NMS) {
        out_scores[(size_t)b * POST_NMS + rank] = ssc[j];
        const float* row = pb + (size_t)sidx[j] * 7;
        float* o = out_boxes + ((size_t)b * POST_NMS + rank) * 7;
        #pragma unroll
        for (int c2 = 0; c2 < 7; ++c2) o[c2] = row[c2];
      }
    }
  }
}

// ---------------- host launcher ----------------
extern "C" void kernel_launch(void* const* d_in, const int* in_sizes, int n_in,
                              void* d_out, int out_size, void* d_ws, size_t ws_size,
                              hipStream_t stream) {
  const float* rpn_cls    = (const float*)d_in[0];  // (B,N)
  const float* rpn_reg    = (const float*)d_in[1];  // (B,N,76)
  const float* xyz        = (const float*)d_in[2];  // (B,N,3)
  const float* mean_sizes = (const float*)d_in[3];  // (1,3)

  // d_out: bbox3d (B,512,7) then scores (B,512)
  float* out_boxes  = (float*)d_out;
  float* out_scores = out_boxes + (size_t)BATCH * POST_NMS * 7;

  // workspace: props (B,N,7) | sorted scores (B,N) | sorted idx (B,N)
  float* props    = (float*)d_ws;
  float* s_scores = props + (size_t)BATCH * NPTS * 7;
  int*   s_idx    = (int*)(s_scores + (size_t)BATCH * NPTS);

  decode_kernel<<<(BATCH * NPTS) / 256, 256, 0, stream>>>(rpn_reg, xyz, mean_sizes, props);
  sort_kernel  <<<BATCH, 1024, 0, stream>>>(rpn_cls, s_scores, s_idx);
  nms_kernel   <<<BATCH, 1024, 0, stream>>>(props, s_idx, s_scores, out_boxes, out_scores);
}